// LutLayer_47648367181962
// MI455X (gfx1250) — compile-verified
//
#include <hip/hip_runtime.h>

// CDNA5 / gfx1250: wave32, WMMA f32 16x16x4 (A=16x4 f32 in 2 VGPRs, B=4x16, C/D=8 VGPRs)
typedef float v2f __attribute__((ext_vector_type(2)));
typedef float v8f __attribute__((ext_vector_type(8)));

#define LUT_EPS 1e-7f

// Hardware transcendentals: v_log_f32 is log2, v_exp_f32 is exp2 — matches the
// base-2 log-domain formulation exactly.
__device__ __forceinline__ float hw_log2(float v) { return __builtin_amdgcn_logf(v); }
__device__ __forceinline__ float hw_exp2(float v) { return __builtin_amdgcn_exp2f(v); }

// 16 rows per wave, 8 waves (256 threads) per block -> 128 rows/block.
__global__ __launch_bounds__(256, 2)
void lutlayer_wmma_kernel(const float* __restrict__ x,      // (rows, 6)
                          const float* __restrict__ lut,    // (depth, 64)
                          const float* __restrict__ table,  // (12, 64) p_q_2_lut_table
                          float* __restrict__ out,          // (rows)
                          int rows, int depth, int dmask)   // dmask = depth-1 if pow2 else -1
{
    // exp2'd 16x16 tile per wave; row stride 20 floats (80B) keeps 16B alignment
    // for ds_load_b128 and is bank-conflict free across the 16 rows.
    __shared__ float tileLDS[8][16 * 20];

    const int tid  = threadIdx.x;
    const int wave = tid >> 5;
    const int lane = tid & 31;
    const int half = lane >> 4;   // 0: lanes 0-15, 1: lanes 16-31
    const int col  = lane & 15;

    const int row_base = blockIdx.x * 128 + wave * 16;

    // ---------------------------------------------------------------
    // Build the custom 12x16 B operand (3 K=4 chunks).
    //  cols 0..7  = "hi" 3-bit patterns (bits j=0..2 of the 6-bit code)
    //  cols 8..15 = "lo" 3-bit patterns (bits j=3..5)
    // Entries come from p_q_2_lut_table with the other half's K rows zeroed,
    // so that c[row, h] + c[row, 8+l] == full log-sum of pattern 8h+l.
    // B f32 4x16 layout: lanes 0-15 -> K = kb,kb+1 ; lanes 16-31 -> K = kb+2,kb+3.
    // ---------------------------------------------------------------
    v2f bmat[3];
    {
        const bool hi   = (col < 8);
        const int  cidx = hi ? (col * 8) : (col - 8);
#pragma unroll
        for (int c = 0; c < 3; ++c) {
            const int k0 = c * 4 + half * 2;
            const int k1 = k0 + 1;
            const bool u0 = hi ? ((k0 < 3) || (k0 >= 6 && k0 < 9))
                               : ((k0 >= 3 && k0 < 6) || (k0 >= 9));
            const bool u1 = hi ? ((k1 < 3) || (k1 >= 6 && k1 < 9))
                               : ((k1 >= 3 && k1 < 6) || (k1 >= 9));
            const float t0 = table[k0 * 64 + cidx];
            const float t1 = table[k1 * 64 + cidx];
            bmat[c].x = u0 ? t0 : 0.0f;
            bmat[c].y = u1 ? t1 : 0.0f;
        }
    }

    // ---------------------------------------------------------------
    // A operand: each lane supplies the 6 log2(p_q) values its K slots need.
    //  chunk c, half h -> p_q indices k0 = 4c + 2h, k0+1
    //  p_q[j] = relu(1 - x_j) + eps  (j < 6)   [the "q" slot]
    //         = relu(x_{j-6}) + eps  (j >= 6)  [the "p" slot]
    // Across the two lane halves every one of the 12 logs is computed once.
    // ---------------------------------------------------------------
    v8f acc = {};
    {
        int r = row_base + col;
        r = (r < rows) ? r : (rows - 1);      // clamp: keep EXEC all-1s for WMMA
        const float* xr = x + (size_t)r * 6;
        const float x0 = xr[0], x1 = xr[1], x2 = xr[2];
        const float x3 = xr[3], x4 = xr[4], x5 = xr[5];

        // chunk0: K 0,1 (q0,q1) | K 2,3 (q2,q3)
        const float s00 = half ? x2 : x0;
        const float s01 = half ? x3 : x1;
        // chunk1: K 4,5 (q4,q5) | K 6,7 (p0,p1)
        const float s10 = half ? x0 : x4;
        const float s11 = half ? x1 : x5;
        // chunk2: K 8,9 (p2,p3) | K 10,11 (p4,p5)
        const float s20 = half ? x4 : x2;
        const float s21 = half ? x5 : x3;

        const float u10 = half ? s10 : (1.0f - s10);
        const float u11 = half ? s11 : (1.0f - s11);

        v2f a0, a1, a2;
        a0.x = hw_log2(fmaxf(1.0f - s00, 0.0f) + LUT_EPS);
        a0.y = hw_log2(fmaxf(1.0f - s01, 0.0f) + LUT_EPS);
        a1.x = hw_log2(fmaxf(u10, 0.0f) + LUT_EPS);
        a1.y = hw_log2(fmaxf(u11, 0.0f) + LUT_EPS);
        a2.x = hw_log2(fmaxf(s20, 0.0f) + LUT_EPS);
        a2.y = hw_log2(fmaxf(s21, 0.0f) + LUT_EPS);

        // K=12 contraction as three accumulated 16x16x4 f32 WMMAs.
        acc = __builtin_amdgcn_wmma_f32_16x16x4_f32(false, a0, false, bmat[0],
                                                    (short)0, acc, false, false);
        acc = __builtin_amdgcn_wmma_f32_16x16x4_f32(false, a1, false, bmat[1],
                                                    (short)0, acc, false, false);
        acc = __builtin_amdgcn_wmma_f32_16x16x4_f32(false, a2, false, bmat[2],
                                                    (short)0, acc, false, false);

        // exp2 the log-sums (16 per row instead of 64) and stage the 16x16
        // tile to LDS for the cross-lane reshape.
        float* tl = tileLDS[wave];
#pragma unroll
        for (int v = 0; v < 8; ++v)
            tl[(v + half * 8) * 20 + col] = hw_exp2(acc[v]);
    }

    __syncthreads();

    // ---------------------------------------------------------------
    // Weighted reduction: lane = (half, r). out[row r] =
    //   sum_h  Ehi[h] * sum_l lut[d, 8h+l] * Elo[l]
    // Each half handles 4 of the 8 hi patterns; one shfl_xor(16) merges.
    // lut row streamed from L2 (1 MB total, fully L2-resident, 512x reuse).
    // ---------------------------------------------------------------
    {
        const int r    = col;
        const int grow = row_base + r;
        const bool valid = grow < rows;
        const int gr   = valid ? grow : (rows - 1);
        // row -> depth index; single v_and when depth is a power of two
        const int d    = (dmask >= 0) ? (gr & dmask) : (gr % depth);
        const float* wrow = lut + (size_t)d * 64 + half * 32;
        const float* tl   = tileLDS[wave] + r * 20;

        float elo[8], ehi[4];
#pragma unroll
        for (int l = 0; l < 8; ++l) elo[l] = tl[8 + l];          // ds_load_b128 x2
#pragma unroll
        for (int h = 0; h < 4; ++h) ehi[h] = tl[half * 4 + h];   // ds_load_b128

        float racc = 0.0f;
#pragma unroll
        for (int h = 0; h < 4; ++h) {
            float s = 0.0f;
#pragma unroll
            for (int l = 0; l < 8; ++l)
                s = fmaf(wrow[h * 8 + l], elo[l], s);
            racc = fmaf(ehi[h], s, racc);
        }
        racc += __shfl_xor(racc, 16, 32);
        if (half == 0 && valid) out[grow] = racc;   // 64B coalesced per wave
    }
}

extern "C" void kernel_launch(void* const* d_in, const int* in_sizes, int n_in,
                              void* d_out, int out_size, void* d_ws, size_t ws_size,
                              hipStream_t stream)
{
    const float* x     = (const float*)d_in[0];   // inputs (B, depth, 6) f32
    const float* lut   = (const float*)d_in[1];   // lut (depth, 64) f32
    const float* table = (const float*)d_in[2];   // p_q_2_lut_table (12, 64) f32
    float* out = (float*)d_out;

    const long long rows = (long long)in_sizes[0] / 6;   // B * depth
    const int depth = in_sizes[1] / 64;
    const int dmask = (depth > 0 && (depth & (depth - 1)) == 0) ? (depth - 1) : -1;
    const int blocks = (int)((rows + 127) / 128);

    lutlayer_wmma_kernel<<<blocks, 256, 0, stream>>>(x, lut, table, out,
                                                     (int)rows, depth, dmask);
    (void)n_in; (void)out_size; (void)d_ws; (void)ws_size;
}